// NCKARTLayer_10728828305803
// MI455X (gfx1250) — compile-verified
//
#include <hip/hip_runtime.h>

typedef __attribute__((ext_vector_type(2))) float v2f;
typedef __attribute__((ext_vector_type(4))) float v4f;
typedef __attribute__((ext_vector_type(8))) float v8f;

#define BATCH   96
#define OUTF    96
#define GPO     8            // GRID_PLUS_ORDER
#define PSTRIDE (BATCH*OUTF) // 9216 elements per j-plane of P

// ---------------------------------------------------------------------------
// Kernel 1: P[j][b][o] = sum_i x[b,i] * W[o,i,j]   for j = 0..2
// One wave32 computes one 16x16 output tile with V_WMMA_F32_16X16X4_F32
// (K dimension = 3, padded to 4 with zeros).
// ---------------------------------------------------------------------------
__global__ void kan_project_wmma(const float* __restrict__ x,
                                 const float* __restrict__ w,
                                 float* __restrict__ P) {
  const int bt   = blockIdx.x;      // batch tile  (M)
  const int ot   = blockIdx.y;      // out-feature tile (N)
  const int j    = blockIdx.z;      // spline coefficient index (0..2)
  const int lane = threadIdx.x;     // wave32, EXEC all-ones
  const int half = lane >> 4;       // 0 -> K={0,1}, 1 -> K={2,3}
  const int l16  = lane & 15;

  // A matrix 16x4 f32 (M x K), M = batch rows.
  // Layout: lanes 0-15 hold K=0 (v.x), K=1 (v.y); lanes 16-31 hold K=2, K=3.
  const int m = bt * 16 + l16;
  v2f a;
  a.x = half ? x[m * 3 + 2] : x[m * 3 + 0];
  a.y = half ? 0.0f         : x[m * 3 + 1];   // K=3 is zero padding

  // B matrix 4x16 f32 (K x N), N = out features; B[k][n] = W[n, k, j].
  // Mirrored layout: lanes 0-15 hold K=0,1; lanes 16-31 hold K=2,3.
  const int n = ot * 16 + l16;
  v2f b;
  b.x = half ? w[n * 3 * GPO + 2 * GPO + j] : w[n * 3 * GPO + 0 * GPO + j];
  b.y = half ? 0.0f                         : w[n * 3 * GPO + 1 * GPO + j];

  v8f c = {};
  // 8 args: (neg_a, A, neg_b, B, c_mod, C, reuse_a, reuse_b)
  c = __builtin_amdgcn_wmma_f32_16x16x4_f32(false, a, false, b,
                                            (short)0, c, false, false);

  // D 16x16 f32: VGPR r -> M = r (lanes 0-15) / r+8 (lanes 16-31), N = lane%16
  float* Pj = P + j * PSTRIDE;
#pragma unroll
  for (int r = 0; r < 8; ++r) {
    const int brow = bt * 16 + r + half * 8;
    Pj[brow * OUTF + n] = c[r];
  }
}

// ---------------------------------------------------------------------------
// Kernel 2: out[b,o1,o2,o3] = P0[b,o1] * P1[b,o2] * P2[b,o3]
// One block per (b,o1): writes a contiguous 96*96 = 9216-float (36 KB) row.
// 340 MB total stream >> 192 MB L2  ->  non-temporal 128-bit stores.
// ---------------------------------------------------------------------------
__global__ void kan_expand(const float* __restrict__ P,
                           float* __restrict__ out) {
  const int bid = blockIdx.x;       // b*96 + o1, 9216 blocks
  const int b   = bid / OUTF;
  const int o1  = bid % OUTF;
  const int tid = threadIdx.x;      // 256 threads = 8 waves

  __shared__ float p1[OUTF];
  __shared__ float p2[OUTF];
  if (tid < OUTF) {
    p1[tid] = P[1 * PSTRIDE + b * OUTF + tid];
  } else if (tid >= 128 && tid < 128 + OUTF) {
    p2[tid - 128] = P[2 * PSTRIDE + b * OUTF + (tid - 128)];
  }
  const float s = P[0 * PSTRIDE + b * OUTF + o1];
  __syncthreads();

  float* dst = out + (size_t)bid * (size_t)(OUTF * OUTF);

  // 9216 floats = 2304 float4; 256 threads x 9 iterations, fully coalesced.
#pragma unroll
  for (int it = 0; it < 9; ++it) {
    const int t  = tid + it * 256;      // 0..2303
    const int o2 = t / 24;              // 24 float4 per o2 row
    const int q  = t % 24;
    const float v = s * p1[o2];
    const float* p2q = &p2[q * 4];
    v4f r;
    r.x = v * p2q[0];
    r.y = v * p2q[1];
    r.z = v * p2q[2];
    r.w = v * p2q[3];
    __builtin_nontemporal_store(r, (v4f*)(dst + o2 * OUTF + q * 4));
  }
}

// ---------------------------------------------------------------------------
extern "C" void kernel_launch(void* const* d_in, const int* in_sizes, int n_in,
                              void* d_out, int out_size, void* d_ws, size_t ws_size,
                              hipStream_t stream) {
  const float* x = (const float*)d_in[0];   // (96, 3)   f32
  const float* w = (const float*)d_in[1];   // (96, 3, 8) f32
  float* outp    = (float*)d_out;           // (96,96,96,96) f32
  float* P       = (float*)d_ws;            // 3 * 9216 f32 = 110,592 B scratch

  // Projection: 6 x 6 tiles of 16, 3 coefficient planes; one wave32 per tile.
  dim3 g1(BATCH / 16, OUTF / 16, 3);
  kan_project_wmma<<<g1, 32, 0, stream>>>(x, w, P);

  // Expansion: one block per (b, o1) row, streaming NT stores.
  dim3 g2(BATCH * OUTF);
  kan_expand<<<g2, 256, 0, stream>>>(P, outp);
}